// BFPQuantizer_29789893165176
// MI455X (gfx1250) — compile-verified
//
#include <hip/hip_runtime.h>
#include <stdint.h>

// ---------------------------------------------------------------------------
// BFP16 (8-bit shared exponent + sign/7-bit mantissa) block quantizer,
// BLOCK_SIZE=16 along the contiguous axis. Pure streaming op:
//   512 MB traffic / 23.3 TB/s  =>  ~22 us floor on MI455X.
// wave32: 4 lanes per 16-float block (float4/lane, 512B/wave contiguous per
// instruction). Block max via two DPP quad_perm exchanges (pure VALU, no DS).
// Double-buffered async global->LDS pipeline (global_load_async_to_lds_b128
// + s_wait_asynccnt). Full steps are branch-free; one guarded tail step max.
// ---------------------------------------------------------------------------

typedef float v4f __attribute__((ext_vector_type(4)));

#define WG_THREADS   256
#define F_PER_STEP   8                              // float4s per thread per superstep
#define CHUNK_BYTES  (WG_THREADS * 16)              // 4 KiB per chunk per WG
#define CHUNK_FLOATS (CHUNK_BYTES / 4)              // 1024
#define STEP_BYTES   (CHUNK_BYTES * F_PER_STEP)     // 32 KiB per superstep per WG
#define STEP_FLOATS  (STEP_BYTES / 4)               // 8192 floats
#define M_SHIFT      6                              // M_BITS-2 with M_BITS=8
#define QMAX         127.0f                         // 2^(M_BITS-1)-1

#define DPP_QUAD_XOR1 0xB1   // quad_perm:[1,0,3,2]
#define DPP_QUAD_XOR2 0x4E   // quad_perm:[2,3,0,1]

__device__ __forceinline__ uint32_t lds_byte_off(const void* p) {
    // Flat shared-aperture address: low 32 bits are the LDS byte offset
    // (ISA 10.2).
    return (uint32_t)(uintptr_t)p;
}

__device__ __forceinline__ int shared_exp_of(float m) {
#if __has_builtin(__builtin_amdgcn_frexp_expf)
    // v_frexp_exp_i32_f32: m = f*2^e, f in [0.5,1)  ->  shared_exp = e-1.
    // frexp_exp(0) == 0 -> e-1 = -1; harmless (all-zero block quantizes to 0).
    return __builtin_amdgcn_frexp_expf(m) - 1;
#else
    union { float f; uint32_t u; } c; c.f = m;
    return (int)((c.u >> 23) & 0xFF) - 127;  // normal-number fallback
#endif
}

// max with the DPP quad-permuted value (cross-lane exchange inside each
// aligned 4-lane group; single VALU op, no DS pipe).
template <int CTRL>
__device__ __forceinline__ float quad_max(float m) {
    int i = __float_as_int(m);
    int p = __builtin_amdgcn_update_dpp(i, i, CTRL, 0xF, 0xF, true);
    return fmaxf(m, __int_as_float(p));
}

// --- issue: exactly 8 wave-wide async copies, no guards (full step) ---------
__device__ __forceinline__ void issue_full(const float* __restrict__ base,
                                           uint32_t lbase, uint32_t goff) {
#pragma unroll
    for (int f = 0; f < F_PER_STEP; ++f) {
        asm volatile("global_load_async_to_lds_b128 %0, %1, %2"
                     :
                     : "v"(lbase + (uint32_t)(f * CHUNK_BYTES)),
                       "v"(goff + (uint32_t)(f * CHUNK_BYTES)),
                       "s"(base)
                     : "memory");
    }
}

// --- issue: guarded (final partial step only) -------------------------------
__device__ __forceinline__ void issue_tail(const float* __restrict__ base,
                                           int rem_floats,
                                           uint32_t lbase, uint32_t goff, int tid) {
    const int lf = tid * 4;
#pragma unroll
    for (int f = 0; f < F_PER_STEP; ++f) {
        if (lf + f * CHUNK_FLOATS < rem_floats) {
            asm volatile("global_load_async_to_lds_b128 %0, %1, %2"
                         :
                         : "v"(lbase + (uint32_t)(f * CHUNK_BYTES)),
                           "v"(goff + (uint32_t)(f * CHUNK_BYTES)),
                           "s"(base)
                         : "memory");
        }
    }
}

// --- one float4's worth of quantization -------------------------------------
__device__ __forceinline__ float qelem(float v, float inv, float stp) {
    // clip(rint(v*inv), -127, 127) * stp   (v_med3_f32 for the clamp)
    return __builtin_amdgcn_fmed3f(__builtin_rintf(v * inv), -QMAX, QMAX) * stp;
}

__device__ __forceinline__ void quant4(const char* lds_buf, float* __restrict__ ybase,
                                       int f, int tid) {
    const v4f xv = *(const v4f*)(lds_buf + f * CHUNK_BYTES + tid * 16);
    // lane-local max|x| over 4 elems, then block max across the 4-lane group
    float m = fmaxf(fmaxf(__builtin_fabsf(xv.x), __builtin_fabsf(xv.y)),
                    fmaxf(__builtin_fabsf(xv.z), __builtin_fabsf(xv.w)));
    m = quad_max<DPP_QUAD_XOR1>(m);
    m = quad_max<DPP_QUAD_XOR2>(m);
    const int   se  = shared_exp_of(m);
    const float inv = ldexpf(1.0f, M_SHIFT - se);   // 2^(6-e)
    const float stp = ldexpf(1.0f, se - M_SHIFT);   // 2^(e-6)
    v4f q;
    q.x = qelem(xv.x, inv, stp);
    q.y = qelem(xv.y, inv, stp);
    q.z = qelem(xv.z, inv, stp);
    q.w = qelem(xv.w, inv, stp);
    __builtin_nontemporal_store(q, (v4f*)(ybase + f * CHUNK_FLOATS + tid * 4));
}

// --- process: branch-free (full step) ---------------------------------------
__device__ __forceinline__ void process_full(const char* lds_buf,
                                             float* __restrict__ ybase, int tid) {
#pragma unroll
    for (int f = 0; f < F_PER_STEP; ++f)
        quant4(lds_buf, ybase, f, tid);
}

// --- process: guarded (final partial step only) -----------------------------
__device__ __forceinline__ void process_tail(const char* lds_buf,
                                             float* __restrict__ ybase,
                                             int rem_floats, int tid) {
    const int lf = tid * 4;
#pragma unroll
    for (int f = 0; f < F_PER_STEP; ++f) {
        // n%16==0 => predicate uniform across each 4-lane block group
        if (lf + f * CHUNK_FLOATS < rem_floats)
            quant4(lds_buf, ybase, f, tid);
    }
}

__global__ void __launch_bounds__(WG_THREADS)
bfp16_quant_kernel(const float* __restrict__ x, float* __restrict__ y,
                   int nfull, int nsteps, int spw, int tail_floats) {
    __shared__ __align__(16) char lds[2][STEP_BYTES];  // 64 KiB double buffer
    const int tid = threadIdx.x;

    const int s0 = (int)blockIdx.x * spw;
    int s1 = s0 + spw;
    if (s1 > nsteps) s1 = nsteps;
    if (s0 >= s1) return;

    const uint32_t goff   = (uint32_t)(tid * 16);               // step-local bytes
    const uint32_t lbase0 = lds_byte_off(&lds[0][0]) + goff;
    const uint32_t lbase1 = lds_byte_off(&lds[1][0]) + goff;

    int buf = 0;
    {
        const float* b0 = x + (long long)s0 * STEP_FLOATS;
        if (s0 < nfull) issue_full(b0, lbase0, goff);
        else            issue_tail(b0, tail_floats, lbase0, goff, tid);
    }

    for (int s = s0; s < s1; ++s) {
        const bool have_next = (s + 1 < s1);
        const bool next_full = (s + 1 < nfull);
        if (have_next) {
            const float* bn = x + (long long)(s + 1) * STEP_FLOATS;
            const uint32_t lb = buf ? lbase0 : lbase1;
            if (next_full) issue_full(bn, lb, goff);
            else           issue_tail(bn, tail_floats, lb, goff, tid);
        }
        if (have_next && next_full) {
            // <=8 outstanding => the older superstep's 8 copies retired
            // (async loads complete in order within a wave).
            asm volatile("s_wait_asynccnt 0x8" ::: "memory");
        } else {
            asm volatile("s_wait_asynccnt 0x0" ::: "memory");
        }
        float* yb = y + (long long)s * STEP_FLOATS;
        if (s < nfull) process_full((const char*)lds[buf], yb, tid);
        else           process_tail((const char*)lds[buf], yb, tail_floats, tid);
        buf ^= 1;
    }
}

extern "C" void kernel_launch(void* const* d_in, const int* in_sizes, int n_in,
                              void* d_out, int out_size, void* d_ws, size_t ws_size,
                              hipStream_t stream) {
    (void)n_in; (void)out_size; (void)d_ws; (void)ws_size;
    const float* x = (const float*)d_in[0];
    float*       y = (float*)d_out;
    const long long n  = (long long)in_sizes[0];                 // 8192*8192
    const int nfull    = (int)(n / STEP_FLOATS);                 // 8192 full steps
    const int tail     = (int)(n % STEP_FLOATS);                 // 0 here
    const int nsteps   = nfull + (tail ? 1 : 0);
    int g = nsteps < 2048 ? nsteps : 2048;
    if (g < 1) g = 1;
    const int spw = (nsteps + g - 1) / g;                        // host-side divide
    bfp16_quant_kernel<<<dim3((uint32_t)g), dim3(WG_THREADS), 0, stream>>>(
        x, y, nfull, nsteps, spw, tail);
}